// Model_90675349553325
// MI455X (gfx1250) — compile-verified
//
#include <hip/hip_runtime.h>
#include <hip/hip_bf16.h>
#include <stdint.h>

// Problem constants (match reference: B=4096, E=8, K=4, P=96, C=32)
#define BATCH 4096
#define NEXP  8
#define TOPK  4
#define ROW   3072            // P*C floats per row
#define ROWB  12288           // bytes per row
#define NTOT  (BATCH * TOPK)  // 16384 stitched rows
#define EPSV  2.220446049250313e-16f
#define LOG2E 1.4426950408889634f
#define LN2   0.6931471805599453f

typedef float f32x4 __attribute__((ext_vector_type(4)));

// Native CDNA5 TRANS-pipe ops (v_exp_f32 = exp2, v_log_f32 = log2).
__device__ __forceinline__ float hw_exp2(float x) {
  return __builtin_amdgcn_exp2f(x);
}
__device__ __forceinline__ float hw_log2(float x) {
  return __builtin_amdgcn_logf(x);
}

// ---------------------------------------------------------------------------
// CDNA5 async global->LDS copy: each lane moves 16 bytes (GVS mode).
//   copies 16 bytes from global sbase + voff_bytes to the LDS byte address lds_byte_addr
// Non-temporal: x is streamed exactly once, keep it out of L2 residency.
// Tracked by ASYNCcnt; drained with s_wait_asynccnt.
// ---------------------------------------------------------------------------
__device__ __forceinline__ void async_ld16(unsigned lds_byte_addr,
                                           const float* sbase,
                                           unsigned voff_bytes) {
  asm volatile("global_load_async_to_lds_b128 %0, %1, %2 offset:0 th:TH_LOAD_NT"
               :
               : "v"(lds_byte_addr), "v"(voff_bytes), "s"(sbase)
               : "memory");
}

__device__ __forceinline__ void wait_async0() {
  asm volatile("s_wait_asynccnt 0" ::: "memory");
}

// ---------------------------------------------------------------------------
// Phase 1: invert the expert-grouped dispatch, branch-free, no atomics.
// Each batch row has exactly TOPK positive gates (softmax of top-k), so the
// slot of entry n inside batch b is the rank of its expert among b's
// positive gates. Gate row is loaded unconditionally as 2x float4 so the
// rank computation is pure VALU (cmp+select), no exec-mask branching.
// ---------------------------------------------------------------------------
__global__ __launch_bounds__(256) void build_map_kernel(
    const int* __restrict__ batch_index,
    const int* __restrict__ expert_index,
    const float* __restrict__ gates,
    int* __restrict__ src,
    float* __restrict__ w) {
  int n = blockIdx.x * 256 + threadIdx.x;
  if (n >= NTOT) return;
  int b = batch_index[n];
  int e = expert_index[n];

  const f32x4* grow = (const f32x4*)(gates + (size_t)b * NEXP);
  f32x4 g0 = grow[0];
  f32x4 g1 = grow[1];
  float g[NEXP] = {g0.x, g0.y, g0.z, g0.w, g1.x, g1.y, g1.z, g1.w};

  int slot = 0;
#pragma unroll
  for (int q = 0; q < NEXP; ++q) {
    slot += ((q < e) && (g[q] > 0.0f)) ? 1 : 0;
  }
  src[b * TOPK + slot] = n;
  w[b * TOPK + slot]   = g[e];
}

// ---------------------------------------------------------------------------
// Phase 2: one block (256 threads = 8 waves) per output batch row.
//   - async-copy the 4 gathered source rows (4 x 12KB = 48KB) into LDS
//     (8 waves x 32 lanes x 16B = 4KB per sweep; 12 sweeps cover 48KB)
//   - out[b, j] = ln( sum_r exp2(x_r[j]*log2e + log2(w_r)) ), 0 -> eps
// 48KB LDS -> 6 concurrent workgroups per WGP (320KB pool).
// ---------------------------------------------------------------------------
__global__ __launch_bounds__(256) void moe_combine_kernel(
    const float* __restrict__ x,
    const int* __restrict__ src,
    const float* __restrict__ w,
    float* __restrict__ out) {
  __shared__ f32x4 smem[TOPK * (ROW / 4)];  // 48 KB: 4 rows x 768 float4

  const int b = blockIdx.x;
  const int t = threadIdx.x;

  // 32-bit LDS byte address of the staging buffer (addrspace(3) offset).
  const unsigned lbase =
      (unsigned)(uintptr_t)(__attribute__((address_space(3))) void*)smem;

  float        l2w[TOPK];  // log2(gate weight), folded into the exponent
  const float* rb[TOPK];
#pragma unroll
  for (int r = 0; r < TOPK; ++r) {
    int n  = src[b * TOPK + r];
    l2w[r] = hw_log2(w[b * TOPK + r]);
    rb[r]  = x + (size_t)n * ROW;
  }

  // Issue all async copies (12 per wave), then drain and barrier.
#pragma unroll
  for (int r = 0; r < TOPK; ++r) {
#pragma unroll
    for (int s = 0; s < 3; ++s) {
      unsigned chunk = (unsigned)(s * 256 + t) * 16u;  // byte offset in row
      async_ld16(lbase + (unsigned)r * (unsigned)ROWB + chunk, rb[r], chunk);
    }
  }
  wait_async0();
  __syncthreads();

  const f32x4* s4 = (const f32x4*)smem;
  f32x4*       o4 = (f32x4*)out;

#pragma unroll
  for (int it = 0; it < 3; ++it) {
    int   j  = it * 256 + t;  // 0..767 float4 slots of this batch row
    f32x4 a0 = s4[0 * (ROW / 4) + j];
    f32x4 a1 = s4[1 * (ROW / 4) + j];
    f32x4 a2 = s4[2 * (ROW / 4) + j];
    f32x4 a3 = s4[3 * (ROW / 4) + j];

    f32x4 r;
    r.x = hw_exp2(fmaf(a0.x, LOG2E, l2w[0])) + hw_exp2(fmaf(a1.x, LOG2E, l2w[1])) +
          hw_exp2(fmaf(a2.x, LOG2E, l2w[2])) + hw_exp2(fmaf(a3.x, LOG2E, l2w[3]));
    r.y = hw_exp2(fmaf(a0.y, LOG2E, l2w[0])) + hw_exp2(fmaf(a1.y, LOG2E, l2w[1])) +
          hw_exp2(fmaf(a2.y, LOG2E, l2w[2])) + hw_exp2(fmaf(a3.y, LOG2E, l2w[3]));
    r.z = hw_exp2(fmaf(a0.z, LOG2E, l2w[0])) + hw_exp2(fmaf(a1.z, LOG2E, l2w[1])) +
          hw_exp2(fmaf(a2.z, LOG2E, l2w[2])) + hw_exp2(fmaf(a3.z, LOG2E, l2w[3]));
    r.w = hw_exp2(fmaf(a0.w, LOG2E, l2w[0])) + hw_exp2(fmaf(a1.w, LOG2E, l2w[1])) +
          hw_exp2(fmaf(a2.w, LOG2E, l2w[2])) + hw_exp2(fmaf(a3.w, LOG2E, l2w[3]));

    // ln(x) = ln2 * log2(x), with the reference's 0 -> eps guard.
    r.x = LN2 * hw_log2(r.x == 0.0f ? EPSV : r.x);
    r.y = LN2 * hw_log2(r.y == 0.0f ? EPSV : r.y);
    r.z = LN2 * hw_log2(r.z == 0.0f ? EPSV : r.z);
    r.w = LN2 * hw_log2(r.w == 0.0f ? EPSV : r.w);

    // Output is also write-once: bypass cache residency with NT store.
    __builtin_nontemporal_store(r, &o4[(size_t)b * (ROW / 4) + j]);
  }
}

// ---------------------------------------------------------------------------
// Inputs (setup_inputs order): x [N,P,C] f32, gates [B,E] f32,
// batch_index [N] int, expert_index [N] int.  Output: [B,P,C] f32.
// ---------------------------------------------------------------------------
extern "C" void kernel_launch(void* const* d_in, const int* in_sizes, int n_in,
                              void* d_out, int out_size, void* d_ws, size_t ws_size,
                              hipStream_t stream) {
  (void)in_sizes; (void)n_in; (void)out_size; (void)ws_size;

  const float* x     = (const float*)d_in[0];
  const float* gates = (const float*)d_in[1];
  const int*   bidx  = (const int*)d_in[2];
  const int*   eidx  = (const int*)d_in[3];
  float*       out   = (float*)d_out;

  // Workspace layout: src[NTOT] int, w[NTOT] float  (128 KB total)
  int*   src = (int*)d_ws;
  float* w   = (float*)(src + NTOT);

  build_map_kernel<<<(NTOT + 255) / 256, 256, 0, stream>>>(bidx, eidx, gates,
                                                           src, w);
  moe_combine_kernel<<<BATCH, 256, 0, stream>>>(x, src, w, out);
}